// CombinedLoss_66486093742616
// MI455X (gfx1250) — compile-verified
//
#include <hip/hip_runtime.h>
#include <hip/hip_bf16.h>

// Focal + MDCA combined loss, single-pass streaming kernel for MI455X (gfx1250).
// B=131072 rows, C=1000 cols, f32. Memory-bound: 524 MB read once (~22.5us @23.3TB/s).
// CDNA5 paths used:
//   - global_load_async_to_lds_b128 double-buffering + s_wait_asynccnt (row staging)
//   - v_wmma_f32_16x16x4_f32 ones-matrix GEMM for the cross-wave column-sum reduce

constexpr int Bsz = 131072;
constexpr int Csz = 1000;

typedef __attribute__((ext_vector_type(2))) float v2f;
typedef __attribute__((ext_vector_type(8))) float v8f;

__device__ __forceinline__ float wave_max(float v) {
#pragma unroll
  for (int k = 16; k > 0; k >>= 1) v = fmaxf(v, __shfl_xor(v, k, 32));
  return v;
}
__device__ __forceinline__ float wave_sum(float v) {
#pragma unroll
  for (int k = 16; k > 0; k >>= 1) v += __shfl_xor(v, k, 32);
  return v;
}

// ws layout (floats): colSum[1024] | counts[1024] | focalSum[1]
__global__ void init_ws_kernel(float* ws) {
  for (int i = threadIdx.x; i < 2052; i += blockDim.x) ws[i] = 0.0f;
}

__global__ void hist_kernel(const int* __restrict__ targets, float* __restrict__ counts) {
  int i = blockIdx.x * blockDim.x + threadIdx.x;
  int stride = gridDim.x * blockDim.x;
  for (; i < Bsz; i += stride) atomicAdd(&counts[targets[i]], 1.0f);
}

// Each wave32 processes rows in a grid-stride loop. Row data (4000 B) is staged
// into LDS with async b128 copies, double-buffered. Lane l owns columns
// 4*l + 128*i + j (i=0..7, j=0..3); the C=1000 tail masks exactly at lane 26.
__global__ __launch_bounds__(256) void softmax_kernel(
    const float* __restrict__ logits, const int* __restrict__ targets,
    float* __restrict__ colSum, float* __restrict__ focalSum) {
  __shared__ float lds[16000];  // 8 waves * 2 buffers * 1000 floats = 64000 B
  __shared__ float sFocal;
  const int lane = threadIdx.x & 31;
  const int wave = threadIdx.x >> 5;
  if (threadIdx.x == 0) sFocal = 0.0f;
  __syncthreads();

  const int wavesPerBlock = blockDim.x >> 5;
  const int gwave = blockIdx.x * wavesPerBlock + wave;
  const int numWaves = gridDim.x * wavesPerBlock;

  const unsigned ldsBaseU = (unsigned)(uintptr_t)(&lds[0]);
  const unsigned myLdsLane = ldsBaseU + (unsigned)(wave * 8000 + lane * 16);

  float4 cacc[8];
#pragma unroll
  for (int i = 0; i < 8; ++i) cacc[i] = float4{0.f, 0.f, 0.f, 0.f};
  float focalAcc = 0.0f;

  // Issue one row's async copy into buffer `buf`: 7 full-wave 512B chunks + 26-lane tail.
  // INST_OFFSET is added to BOTH the global and the LDS address (ISA 15.18).
  auto issue_row = [&](int row, int buf) {
    unsigned long long g =
        (unsigned long long)(const void*)(logits + (size_t)row * Csz) +
        (unsigned long long)(lane * 16);
    unsigned l = myLdsLane + (unsigned)(buf * 4000);
#define ASYNC_LD(OFF)                                                     \
  asm volatile("global_load_async_to_lds_b128 %0, %1, off offset:" OFF    \
               ::"v"(l), "v"(g) : "memory")
    ASYNC_LD("0");
    ASYNC_LD("512");
    ASYNC_LD("1024");
    ASYNC_LD("1536");
    ASYNC_LD("2048");
    ASYNC_LD("2560");
    ASYNC_LD("3072");
    if (lane < 26) { ASYNC_LD("3584"); }  // bytes 3584..3999 (cols 896..999)
#undef ASYNC_LD
  };

  int row = gwave;
  int buf = 0;
  if (row < Bsz) issue_row(row, 0);

  while (row < Bsz) {
    const int nxt = row + numWaves;
    if (nxt < Bsz) {
      issue_row(nxt, buf ^ 1);  // prefetch next row into other buffer
      asm volatile("s_wait_asynccnt 0x8" ::: "memory");  // current 8 done, next 8 in flight
    } else {
      asm volatile("s_wait_asynccnt 0x0" ::: "memory");
    }

    const float* xs = lds + wave * 2000 + buf * 1000;
    const float4* p = reinterpret_cast<const float4*>(xs);
    float4 v[8];
#pragma unroll
    for (int i = 0; i < 8; ++i) v[i] = p[i * 32 + lane];

    const bool tailOk = (lane < 26);
    float m = -3.402823466e38f;
#pragma unroll
    for (int i = 0; i < 7; ++i)
      m = fmaxf(m, fmaxf(fmaxf(v[i].x, v[i].y), fmaxf(v[i].z, v[i].w)));
    if (tailOk) m = fmaxf(m, fmaxf(fmaxf(v[7].x, v[7].y), fmaxf(v[7].z, v[7].w)));
    m = wave_max(m);

    float s = 0.0f;
#pragma unroll
    for (int i = 0; i < 8; ++i) {
      const bool ok = (i < 7) || tailOk;
      v[i].x = ok ? __expf(v[i].x - m) : 0.0f;
      v[i].y = ok ? __expf(v[i].y - m) : 0.0f;
      v[i].z = ok ? __expf(v[i].z - m) : 0.0f;
      v[i].w = ok ? __expf(v[i].w - m) : 0.0f;
      s += (v[i].x + v[i].y) + (v[i].z + v[i].w);
    }
    s = wave_sum(s);
    const float invS = 1.0f / s;
#pragma unroll
    for (int i = 0; i < 8; ++i) {
      cacc[i].x += v[i].x * invS;
      cacc[i].y += v[i].y * invS;
      cacc[i].z += v[i].z * invS;
      cacc[i].w += v[i].w * invS;
    }

    // Focal term for this row (gamma == 2); xs[t] is a broadcast LDS read.
    const int t = targets[row];
    const float xt = xs[t];
    const float logpt = xt - m - __logf(s);
    const float pt = __expf(logpt);
    const float omp = 1.0f - pt;
    focalAcc += -(omp * omp) * logpt;

    buf ^= 1;
    row = nxt;
  }

  // ---- Cross-wave column-sum reduction via WMMA (ones-matrix GEMM) ----
  // Each wave dumps its 1024 partial sums into its own LDS region (plain b128
  // stores, no atomics), then the 8x1024 matrix is reduced over the wave axis
  // with v_wmma_f32_16x16x4_f32: A = ones(16x4), B = 4 waves x 16 cols, f32
  // accumulate. With A all-ones, D[m,n] = sum_k
  // B[k,n] regardless of the physical K-slot mapping, so the B-layout K
  // ambiguity is harmless; N = lane&15 matches between B and D per ISA layout.
  __syncthreads();
  float* myCol = lds + wave * 1024;
#pragma unroll
  for (int i = 0; i < 8; ++i) {
    *reinterpret_cast<float4*>(myCol + 4 * lane + 128 * i) = cacc[i];
  }
  if (lane == 0) atomicAdd(&sFocal, focalAcc);
  __syncthreads();

  const int nsub = lane & 15;
  const int khalf = lane >> 4;  // which K row this lane-half supplies
  for (int grp = wave; grp < 64; grp += 8) {   // 64 groups of 16 columns
    const int n0 = grp * 16;
    v2f a;
    a[0] = 1.0f;
    a[1] = 1.0f;
    v2f b0, b1;
    b0[0] = lds[(khalf ? 1 : 0) * 1024 + n0 + nsub];  // waves 0..3
    b0[1] = lds[(khalf ? 3 : 2) * 1024 + n0 + nsub];
    b1[0] = lds[(khalf ? 5 : 4) * 1024 + n0 + nsub];  // waves 4..7
    b1[1] = lds[(khalf ? 7 : 6) * 1024 + n0 + nsub];
    v8f c = {};
    c = __builtin_amdgcn_wmma_f32_16x16x4_f32(false, a, false, b0, (short)0, c,
                                              false, false);
    c = __builtin_amdgcn_wmma_f32_16x16x4_f32(false, a, false, b1, (short)0, c,
                                              false, false);
    // D row M=0: VGPR0, lanes 0-15 (N = lane). All D rows are identical.
    const int col = n0 + nsub;
    if (lane < 16 && col < Csz) atomicAdd(&colSum[col], c[0]);
  }
  if (threadIdx.x == 0) atomicAdd(focalSum, sFocal);
}

__global__ void finalize_kernel(const float* __restrict__ colSum,
                                const float* __restrict__ counts,
                                const float* __restrict__ focalSum,
                                float* __restrict__ out) {
  __shared__ float red[256];
  float acc = 0.0f;
  for (int c = threadIdx.x; c < Csz; c += blockDim.x)
    acc += fabsf(colSum[c] - counts[c]);
  red[threadIdx.x] = acc;
  __syncthreads();
  for (int s = 128; s > 0; s >>= 1) {
    if ((int)threadIdx.x < s) red[threadIdx.x] += red[threadIdx.x + s];
    __syncthreads();
  }
  if (threadIdx.x == 0) {
    const float invB = 1.0f / (float)Bsz;
    // mdca = mean_c |colSum/B - counts/B| = sumAbs / (B*C)
    out[0] = focalSum[0] * invB + 5.0f * (red[0] * invB * (1.0f / (float)Csz));
  }
}

extern "C" void kernel_launch(void* const* d_in, const int* in_sizes, int n_in,
                              void* d_out, int out_size, void* d_ws, size_t ws_size,
                              hipStream_t stream) {
  const float* logits = (const float*)d_in[0];
  const int* targets = (const int*)d_in[1];
  float* out = (float*)d_out;

  float* colSum = (float*)d_ws;     // [1024]
  float* counts = colSum + 1024;    // [1024]
  float* focal = counts + 1024;     // [1]

  init_ws_kernel<<<1, 256, 0, stream>>>((float*)d_ws);
  hist_kernel<<<256, 256, 0, stream>>>(targets, counts);
  softmax_kernel<<<512, 256, 0, stream>>>(logits, targets, colSum, focal);
  finalize_kernel<<<1, 256, 0, stream>>>(colSum, counts, focal, out);
}